// TriPlane_28114855919973
// MI455X (gfx1250) — compile-verified
//
#include <hip/hip_runtime.h>
#include <stddef.h>
#include <stdint.h>

// Problem geometry (fixed by the reference's setup_inputs)
#define M_PTS   131072          // points (power of two: 2^17)
#define M_LOG2  17
#define NB_IMG  12              // N(4) * n_planes(3)
#define C_CH    32
#define H_IMG   256
#define W_IMG   256
#define HW_IMG  (H_IMG * W_IMG)
#define TROW    36              // padded LDS tile row (16B-aligned rows, stride 144B)

// gfx1250 async global->LDS path (probe-safe fallback chain: b128 -> b32 -> plain)
#if defined(__HIP_DEVICE_COMPILE__) &&                                         \
    __has_builtin(__builtin_amdgcn_global_load_async_to_lds_b32) &&            \
    __has_builtin(__builtin_amdgcn_s_wait_asynccnt)
#define USE_ASYNC_LDS 1
#else
#define USE_ASYNC_LDS 0
#endif

#if USE_ASYNC_LDS && __has_builtin(__builtin_amdgcn_global_load_async_to_lds_b128)
#define USE_ASYNC_LDS_B128 1
#else
#define USE_ASYNC_LDS_B128 0
#endif

#if USE_ASYNC_LDS
typedef __attribute__((address_space(1))) int g_i32;   // global (AS1) scalar
typedef __attribute__((address_space(3))) int l_i32;   // LDS (AS3) scalar
typedef int v4i_t __attribute__((vector_size(16)));    // 4 x i32
typedef __attribute__((address_space(1))) v4i_t g_v4i; // global (AS1) vector
typedef __attribute__((address_space(3))) v4i_t l_v4i; // LDS (AS3) vector
#endif

__device__ __forceinline__ int iclamp(int v, int lo, int hi) {
    return v < lo ? lo : (v > hi ? hi : v);
}

// ---------------------------------------------------------------------------
// Kernel 1: transpose feature maps (12, C=32, H, W) -> (12, H, W, C=32).
// One block handles (b, y, 32-wide x tile): 32x32 f32 tile through LDS.
// Load phase: one async b128 per lane (lane = (c, 4x)): 128B contiguous per
// 8-lane group, 16B-aligned LDS destinations (rows padded to 36 floats).
// Store phase: float4 per lane (lane = (x, 4c)): contiguous b128 stores.
// ---------------------------------------------------------------------------
__global__ void __launch_bounds__(256)
tri_transpose_chw_to_hwc(const float* __restrict__ img, float* __restrict__ outT) {
    __shared__ float tile[C_CH][TROW];

    const int blk = blockIdx.x;           // NB_IMG * H * (W/32) = 24576 blocks
    const int xt  = blk & 7;              // 8 x-tiles of 32
    const int y   = (blk >> 3) & (H_IMG - 1);
    const int b   = blk >> 11;            // [0, 12)
    const int x0  = xt * 32;

    const float* src = img + (size_t)b * C_CH * HW_IMG + (size_t)y * W_IMG + x0;

#if USE_ASYNC_LDS_B128
    {   // one b128 async per lane covers the whole 32x32 tile
        const int c   = (int)threadIdx.x >> 3;        // channel 0..31
        const int xi4 = ((int)threadIdx.x & 7) << 2;  // x within tile: 0,4,...,28
        const float* g = src + (size_t)c * HW_IMG + xi4;      // 16B aligned
        __builtin_amdgcn_global_load_async_to_lds_b128(
            (g_v4i*)g, (l_v4i*)&tile[c][xi4], /*offset=*/0, /*cpol=*/0);
    }
#elif USE_ASYNC_LDS
#pragma unroll
    for (int k = 0; k < 4; ++k) {
        const int e  = (int)threadIdx.x + k * 256;  // 1024 elements per tile
        const int c  = e >> 5;
        const int xi = e & 31;
        const float* g = src + (size_t)c * HW_IMG + xi;
        __builtin_amdgcn_global_load_async_to_lds_b32(
            (g_i32*)g, (l_i32*)&tile[c][xi], /*offset=*/0, /*cpol=*/0);
    }
#else
#pragma unroll
    for (int k = 0; k < 4; ++k) {
        const int e  = (int)threadIdx.x + k * 256;
        const int c  = e >> 5;
        const int xi = e & 31;
        tile[c][xi] = src[(size_t)c * HW_IMG + xi];
    }
#endif

#if USE_ASYNC_LDS
    __builtin_amdgcn_s_wait_asynccnt(0);
#endif
    __syncthreads();

    // Store phase: lane = (xi, c4); float4 of consecutive channels per lane.
    // LDS reads are stride-TROW gathers (worst case 2-way conflict), global
    // writes are fully contiguous b128 per lane.
    {
        const int xi = (int)threadIdx.x >> 3;         // 0..31
        const int c4 = ((int)threadIdx.x & 7) << 2;   // 0,4,...,28
        float4 v;
        v.x = tile[c4 + 0][xi];
        v.y = tile[c4 + 1][xi];
        v.z = tile[c4 + 2][xi];
        v.w = tile[c4 + 3][xi];
        float* dst = outT + ((size_t)b * HW_IMG + (size_t)y * W_IMG + x0 + xi) * C_CH + c4;
        *(float4*)dst = v;
    }
}

// ---------------------------------------------------------------------------
// Shared per-sample bilinear setup.
// Plane projections (from the reference's inverted plane axes):
//   p==0 -> (gx,gy) = (cx, cy)
//   p==1 -> (gx,gy) = (cx, cz)
//   p==2 -> (gx,gy) = (cz, cx)
// with c = (2/1.6) * point = 1.25 * point, unnormalize: u = g*128 + 127.5.
// ---------------------------------------------------------------------------
struct BiSetup {
    int   jx0, jx1, jy0, jy1;     // clamped corner indices
    float w00, w01, w10, w11;     // zero-padded corner weights
};

__device__ __forceinline__ BiSetup bilinear_setup(const float* __restrict__ pts,
                                                  int m, int p) {
    const float px = pts[3 * m + 0];
    const float py = pts[3 * m + 1];
    const float pz = pts[3 * m + 2];
    const float cx = 1.25f * px, cy = 1.25f * py, cz = 1.25f * pz;
    const float gx = (p == 2) ? cz : cx;
    const float gy = (p == 0) ? cy : ((p == 1) ? cz : cx);

    const float x  = fmaf(gx, 128.0f, 127.5f);
    const float y  = fmaf(gy, 128.0f, 127.5f);
    const float xf = floorf(x), yf = floorf(y);
    const float fx = x - xf,    fy = y - yf;
    const int ix0 = (int)xf, iy0 = (int)yf;
    const int ix1 = ix0 + 1, iy1 = iy0 + 1;
    const float wx0 = 1.0f - fx, wy0 = 1.0f - fy;

    const bool vx0 = (unsigned)ix0 < (unsigned)W_IMG;
    const bool vx1 = (unsigned)ix1 < (unsigned)W_IMG;
    const bool vy0 = (unsigned)iy0 < (unsigned)H_IMG;
    const bool vy1 = (unsigned)iy1 < (unsigned)H_IMG;

    BiSetup s;
    s.w00 = (vx0 && vy0) ? wx0 * wy0 : 0.0f;
    s.w01 = (vx1 && vy0) ? fx  * wy0 : 0.0f;
    s.w10 = (vx0 && vy1) ? wx0 * fy  : 0.0f;
    s.w11 = (vx1 && vy1) ? fx  * fy  : 0.0f;
    s.jx0 = iclamp(ix0, 0, W_IMG - 1);
    s.jx1 = iclamp(ix1, 0, W_IMG - 1);
    s.jy0 = iclamp(iy0, 0, H_IMG - 1);
    s.jy1 = iclamp(iy1, 0, H_IMG - 1);
    return s;
}

// ---------------------------------------------------------------------------
// Kernel 2 (fast path): sample from HWC-transposed features in d_ws.
// One thread = (sample, 4 channels). Lane group of 8 covers all 32 channels of
// one sample; each corner fetch is one global_load_b128 (128B/corner/sample).
// Output: float4 store -> contiguous 128B per sample per wave.
// ---------------------------------------------------------------------------
__global__ void __launch_bounds__(256)
tri_sample_hwc(const float* __restrict__ pts, const float* __restrict__ featT,
               float* __restrict__ out) {
    const int t  = (int)blockIdx.x * 256 + (int)threadIdx.x;
    const int s  = t >> 3;                 // sample id: b*M + m
    if (s >= NB_IMG * M_PTS) return;
    const int c4 = (t & 7) << 2;           // channel group base
    const int m  = s & (M_PTS - 1);
    const int b  = s >> M_LOG2;            // [0, 12)
    const int p  = b % 3;

    const BiSetup bs = bilinear_setup(pts, m, p);

    const float* base = featT + (size_t)b * HW_IMG * C_CH;
    const float4 f00 = *(const float4*)(base + ((size_t)(bs.jy0 * W_IMG + bs.jx0) * C_CH + c4));
    const float4 f01 = *(const float4*)(base + ((size_t)(bs.jy0 * W_IMG + bs.jx1) * C_CH + c4));
    const float4 f10 = *(const float4*)(base + ((size_t)(bs.jy1 * W_IMG + bs.jx0) * C_CH + c4));
    const float4 f11 = *(const float4*)(base + ((size_t)(bs.jy1 * W_IMG + bs.jx1) * C_CH + c4));

    float4 r;
    r.x = fmaf(bs.w11, f11.x, fmaf(bs.w10, f10.x, fmaf(bs.w01, f01.x, bs.w00 * f00.x)));
    r.y = fmaf(bs.w11, f11.y, fmaf(bs.w10, f10.y, fmaf(bs.w01, f01.y, bs.w00 * f00.y)));
    r.z = fmaf(bs.w11, f11.z, fmaf(bs.w10, f10.z, fmaf(bs.w01, f01.z, bs.w00 * f00.z)));
    r.w = fmaf(bs.w11, f11.w, fmaf(bs.w10, f10.w, fmaf(bs.w01, f01.w, bs.w00 * f00.w)));

    *(float4*)(out + (size_t)s * C_CH + c4) = r;
}

// ---------------------------------------------------------------------------
// Kernel 2 (fallback): sample directly from CHW layout (if ws too small).
// ---------------------------------------------------------------------------
__global__ void __launch_bounds__(256)
tri_sample_chw(const float* __restrict__ pts, const float* __restrict__ img,
               float* __restrict__ out) {
    const int t  = (int)blockIdx.x * 256 + (int)threadIdx.x;
    const int s  = t >> 3;
    if (s >= NB_IMG * M_PTS) return;
    const int c4 = (t & 7) << 2;
    const int m  = s & (M_PTS - 1);
    const int b  = s >> M_LOG2;
    const int p  = b % 3;

    const BiSetup bs = bilinear_setup(pts, m, p);
    const float* base = img + (size_t)b * C_CH * HW_IMG;

    float r[4];
#pragma unroll
    for (int q = 0; q < 4; ++q) {
        const float* bc = base + (size_t)(c4 + q) * HW_IMG;
        float acc =      bs.w00 * bc[bs.jy0 * W_IMG + bs.jx0];
        acc = fmaf(bs.w01, bc[bs.jy0 * W_IMG + bs.jx1], acc);
        acc = fmaf(bs.w10, bc[bs.jy1 * W_IMG + bs.jx0], acc);
        acc = fmaf(bs.w11, bc[bs.jy1 * W_IMG + bs.jx1], acc);
        r[q] = acc;
    }
    *(float4*)(out + (size_t)s * C_CH + c4) = make_float4(r[0], r[1], r[2], r[3]);
}

// ---------------------------------------------------------------------------
extern "C" void kernel_launch(void* const* d_in, const int* in_sizes, int n_in,
                              void* d_out, int out_size, void* d_ws, size_t ws_size,
                              hipStream_t stream) {
    (void)in_sizes; (void)n_in; (void)out_size;
    const float* points = (const float*)d_in[0];   // (131072, 3) f32
    const float* images = (const float*)d_in[1];   // (4, 3, 32, 256, 256) f32
    float* out = (float*)d_out;                    // (4, 3, 131072, 32) f32

    const size_t feat_bytes = (size_t)NB_IMG * HW_IMG * C_CH * sizeof(float); // ~100.7 MB
    const int sample_threads = NB_IMG * M_PTS * 8;                // 12,582,912
    const int sample_blocks  = (sample_threads + 255) / 256;      // 49,152

    if (ws_size >= feat_bytes) {
        float* featT = (float*)d_ws;
        const int tblocks = NB_IMG * H_IMG * (W_IMG / 32);        // 24,576
        tri_transpose_chw_to_hwc<<<tblocks, 256, 0, stream>>>(images, featT);
        tri_sample_hwc<<<sample_blocks, 256, 0, stream>>>(points, featT, out);
    } else {
        tri_sample_chw<<<sample_blocks, 256, 0, stream>>>(points, images, out);
    }
}